// NeuralNetPrescriptionHistory_61538291417849
// MI455X (gfx1250) — compile-verified
//
#include <hip/hip_runtime.h>
#include <hip/hip_bf16.h>

typedef float v2f __attribute__((ext_vector_type(2)));
typedef float v4f __attribute__((ext_vector_type(4)));
typedef float v8f __attribute__((ext_vector_type(8)));

#define B_VISITS 16384
#define EMB 128
#define XDIM 384
#define HID 64
#define MED 153
#define TILE_M 16

__device__ __forceinline__ int lower_bound_i(const int* __restrict__ a, int n, int key) {
    int lo = 0, hi = n;
    while (lo < hi) {
        int mid = (lo + hi) >> 1;
        if (a[mid] < key) lo = mid + 1; else hi = mid;
    }
    return lo;
}

__device__ __forceinline__ v8f wmma_f32_k4(v2f a, v2f b, v8f c) {
    // V_WMMA_F32_16X16X4_F32 : D = A(16x4) * B(4x16) + C(16x16), all f32
    return __builtin_amdgcn_wmma_f32_16x16x4_f32(
        /*neg_a=*/false, a, /*neg_b=*/false, b,
        /*c_mod=*/(short)0, c, /*reuse_a=*/false, /*reuse_b=*/false);
}

// Fully fused: segment-sum embeddings -> LDS x[16,384] -> WMMA GEMM1 (+bias,ReLU)
// -> LDS h[16,64] -> WMMA GEMM2 (+bias,sigmoid) -> out[16,153].
// One block (8 waves, wave32) per 16-visit tile.
__global__ void __launch_bounds__(256)
fused_visit_mlp(const int* __restrict__ diag_codes, const int* __restrict__ diag_seg, int n_diag,
                const int* __restrict__ proc_codes, const int* __restrict__ proc_seg, int n_proc,
                const int* __restrict__ med_codes,  const int* __restrict__ med_seg,  int n_med,
                const float* __restrict__ diag_table,
                const float* __restrict__ proc_table,
                const float* __restrict__ med_table,
                const float* __restrict__ W1, const float* __restrict__ b1,
                const float* __restrict__ W2, const float* __restrict__ b2,
                float* __restrict__ out) {
    __shared__ alignas(16) float xs[TILE_M * XDIM];   // 24 KB
    __shared__ alignas(16) float hs[TILE_M * HID];    //  4 KB

    const int tid     = threadIdx.x;
    const int wave    = tid >> 5;          // 0..7
    const int lane    = tid & 31;          // 0..31
    const int halfSel = lane >> 4;         // 0|1
    const int laneM   = lane & 15;         // row (A/C) or col (B/C)
    const int rowBase = blockIdx.x * TILE_M;

    // ---------------- Stage 1: embedding segment sums into LDS ----------------
    // 48 (table, visit) pairs; wave w handles pairs [6w, 6w+6).
    for (int p = wave * 6; p < wave * 6 + 6; ++p) {
        const int t = p >> 4;              // 0=diag, 1=proc, 2=med
        const int v = p & 15;              // visit within tile
        const int visit = rowBase + v;

        const int*   codes; const int* seg; const float* table; int n;
        if (t == 0)      { codes = diag_codes; seg = diag_seg; table = diag_table; n = n_diag; }
        else if (t == 1) { codes = proc_codes; seg = proc_seg; table = proc_table; n = n_proc; }
        else             { codes = med_codes;  seg = med_seg;  table = med_table;  n = n_med;  }

        const int lo = lower_bound_i(seg, n, visit);
        const int hi = lower_bound_i(seg, n, visit + 1);

        v4f acc = {0.f, 0.f, 0.f, 0.f};
        for (int i = lo; i < hi; ++i) {
            const int c = codes[i];
            const v4f row = *reinterpret_cast<const v4f*>(table + (size_t)c * EMB + lane * 4);
            acc += row;
        }
        *reinterpret_cast<v4f*>(&xs[v * XDIM + t * EMB + lane * 4]) = acc;
    }
    __syncthreads();

    // ---------------- Stage 2: h = relu(x @ W1 + b1), WMMA f32 16x16x4 --------
    // N = 64 -> 4 waves each own one 16x16 tile of h; waves 4..7 idle here.
    if (wave < 4) {
        const int nBase = wave * 16;
        const int koff  = halfSel * 2;
        v8f c = {0.f, 0.f, 0.f, 0.f, 0.f, 0.f, 0.f, 0.f};
        #pragma unroll 4
        for (int k = 0; k < XDIM; k += 4) {
            // A: lane<16 -> K=k,k+1 ; lane>=16 -> K=k+2,k+3 (row = laneM)
            v2f a = *reinterpret_cast<const v2f*>(&xs[laneM * XDIM + k + koff]);
            // B: same K split, col = laneM
            v2f b;
            b.x = W1[(k + koff)     * HID + nBase + laneM];
            b.y = W1[(k + koff + 1) * HID + nBase + laneM];
            c = wmma_f32_k4(a, b, c);
        }
        const float bias = b1[nBase + laneM];
        #pragma unroll
        for (int r = 0; r < 8; ++r) {
            const int rowInTile = r + halfSel * 8;
            float v = c[r] + bias;
            hs[rowInTile * HID + nBase + laneM] = v > 0.f ? v : 0.f;
        }
    }
    __syncthreads();

    // ---------------- Stage 3: out = sigmoid(h @ W2 + b2) ---------------------
    // N padded to 160 -> 10 tiles of 16 over 8 waves (wave-uniform loop).
    const int koff = halfSel * 2;
    for (int nt = wave; nt < 10; nt += 8) {
        const int nBase = nt * 16;
        const int n     = nBase + laneM;
        const bool nok  = (n < MED);
        v8f c = {0.f, 0.f, 0.f, 0.f, 0.f, 0.f, 0.f, 0.f};
        #pragma unroll 4
        for (int k = 0; k < HID; k += 4) {
            v2f a = *reinterpret_cast<const v2f*>(&hs[laneM * HID + k + koff]);
            v2f b;
            b.x = nok ? W2[(k + koff)     * MED + n] : 0.f;
            b.y = nok ? W2[(k + koff + 1) * MED + n] : 0.f;
            c = wmma_f32_k4(a, b, c);
        }
        const float bias = nok ? b2[n] : 0.f;
        #pragma unroll
        for (int r = 0; r < 8; ++r) {
            const int row = rowBase + r + halfSel * 8;
            if (nok) {
                const float v = c[r] + bias;
                out[(size_t)row * MED + n] = 1.0f / (1.0f + __expf(-v));
            }
        }
    }
}

extern "C" void kernel_launch(void* const* d_in, const int* in_sizes, int n_in,
                              void* d_out, int out_size, void* d_ws, size_t ws_size,
                              hipStream_t stream) {
    const int*   diag_codes = (const int*)d_in[0];
    const int*   diag_seg   = (const int*)d_in[1];
    const int*   proc_codes = (const int*)d_in[2];
    const int*   proc_seg   = (const int*)d_in[3];
    const int*   med_codes  = (const int*)d_in[4];
    const int*   med_seg    = (const int*)d_in[5];
    const float* diag_table = (const float*)d_in[6];
    const float* proc_table = (const float*)d_in[7];
    const float* med_table  = (const float*)d_in[8];
    const float* W1         = (const float*)d_in[9];
    const float* b1         = (const float*)d_in[10];
    const float* W2         = (const float*)d_in[11];
    const float* b2         = (const float*)d_in[12];
    float*       out        = (float*)d_out;

    const int n_diag = in_sizes[0];
    const int n_proc = in_sizes[2];
    const int n_med  = in_sizes[4];

    dim3 grid(B_VISITS / TILE_M);   // 1024 tiles... = 16384/16
    dim3 block(256);                // 8 wave32
    fused_visit_mlp<<<grid, block, 0, stream>>>(
        diag_codes, diag_seg, n_diag,
        proc_codes, proc_seg, n_proc,
        med_codes,  med_seg,  n_med,
        diag_table, proc_table, med_table,
        W1, b1, W2, b2, out);
}